// MutInfoEstimator_66907000537550
// MI455X (gfx1250) — compile-verified
//
#include <hip/hip_runtime.h>
#include <hip/hip_bf16.h>
#include <math.h>

typedef __attribute__((ext_vector_type(2))) float v2f;
typedef __attribute__((ext_vector_type(8))) float v8f;
typedef __attribute__((ext_vector_type(4))) unsigned int u32x4;
typedef __attribute__((ext_vector_type(4))) int i32x4;
typedef __attribute__((ext_vector_type(8))) int i32x8;

#define BATCH 4096
#define NS    8192
#define DX    256
#define DY    128
#define XPAD  258   // x tile LDS row stride (floats): +2 pad -> conflict-free ds_load_b64
#define YPAD  130
#define ROWS_PER_WG 32
#define NBLK  2     // two 16-row M blocks per wave: each B fragment feeds 2 WMMAs
#define NWAVE 8
#define NTHREADS 256
#define NPART (BATCH / ROWS_PER_WG)   // 128 partials

#if defined(__has_builtin)
#if __has_builtin(__builtin_amdgcn_tensor_load_to_lds)
#define HAVE_TDM 1
#else
#define HAVE_TDM 0
#endif
#else
#define HAVE_TDM 0
#endif

__device__ __forceinline__ void lse_update(float& m, float& s, float p) {
    float mn = fmaxf(m, p);
    s = s * __expf(m - mn) + __expf(p - mn);
    m = mn;
}

__device__ __forceinline__ void lse_combine(float& m, float& s, float m2, float s2) {
    float mn = fmaxf(m, m2);
    s = s * __expf(m - mn) + s2 * __expf(m2 - mn);
    m = mn;
}

#if HAVE_TDM
// Tensor Data Mover: DMA a rows x cols f32 tile global->LDS, inserting +2 DWORDs of
// LDS padding after every row (pad_interval_code: 7 => 256 DWORDs, 6 => 128 DWORDs).
__device__ __forceinline__ void tdm_load_tile(const float* gptr, unsigned lds_off,
                                              int rows, int cols, int pad_code) {
    unsigned long long ga = (unsigned long long)(uintptr_t)gptr;

    u32x4 g0;
    g0[0] = 1u;                                     // count=1 valid user descriptor
    g0[1] = lds_off;                                // lds_addr (bytes)
    g0[2] = (unsigned)(ga & 0xFFFFFFFFu);           // global_addr[31:0]
    g0[3] = (unsigned)((ga >> 32) & 0x01FFFFFFu)    // global_addr[56:32]
          | 0x80000000u;                            // type=2 ("image")

    i32x8 g1;
    g1[0] = (int)((2u << 16)                        // data_size = 4 bytes
                | (1u << 20)                        // pad_enable
                | ((unsigned)pad_code << 22)        // pad_interval
                | (1u << 25));                      // pad_amount = 2 DWORDs
    g1[1] = (int)((unsigned)(cols & 0xFFFF) << 16); // tensor_dim0[15:0]
    g1[2] = (int)(((unsigned)cols >> 16)            // tensor_dim0[31:16]
                | ((unsigned)(rows & 0xFFFF) << 16)); // tensor_dim1[15:0]
    g1[3] = (int)((unsigned)(cols & 0xFFFF) << 16); // tile_dim0 (dim1 hi = 0)
    g1[4] = rows;                                   // tile_dim1 (tile_dim2 = 0)
    g1[5] = cols;                                   // tensor_dim0_stride[31:0]
    g1[6] = 0;
    g1[7] = 0;

    i32x4 z4 = {0, 0, 0, 0};
#if __clang_major__ >= 23
    i32x8 z8 = {0, 0, 0, 0, 0, 0, 0, 0};
    __builtin_amdgcn_tensor_load_to_lds(g0, g1, z4, z4, z8, 0);
#else
    __builtin_amdgcn_tensor_load_to_lds(g0, g1, z4, z4, 0);
#endif
}
#endif

__global__ __launch_bounds__(NTHREADS)
void mi_main_kernel(const float* __restrict__ x, const float* __restrict__ y,
                    const float* __restrict__ xs, const float* __restrict__ ys,
                    float* __restrict__ partial) {
    __shared__ float xt[ROWS_PER_WG * XPAD];
    __shared__ float yt[ROWS_PER_WG * YPAD];
    __shared__ float red[NWAVE][ROWS_PER_WG][3][2];  // [wave][row][{xy,x,y}][{m,s}]
    __shared__ float zrow[ROWS_PER_WG];

    const int tid  = threadIdx.x;
    const int lane = tid & 31;
    const int wave = tid >> 5;
    const int row0 = blockIdx.x * ROWS_PER_WG;

    // ---- stage 32x256 x-tile and 32x128 y-tile into padded LDS ----
#if HAVE_TDM
    if (wave == 0) {
        tdm_load_tile(x + (size_t)row0 * DX, (unsigned)(uintptr_t)(void*)xt,
                      ROWS_PER_WG, DX, 7);
        tdm_load_tile(y + (size_t)row0 * DY, (unsigned)(uintptr_t)(void*)yt,
                      ROWS_PER_WG, DY, 6);
        __builtin_amdgcn_s_wait_tensorcnt(0);
    }
#else
    for (int idx = tid; idx < ROWS_PER_WG * (DX / 2); idx += NTHREADS) {
        int r = idx / (DX / 2), c = idx % (DX / 2);
        float2 v = *(const float2*)(x + (size_t)(row0 + r) * DX + c * 2);
        xt[r * XPAD + c * 2] = v.x;
        xt[r * XPAD + c * 2 + 1] = v.y;
    }
    for (int idx = tid; idx < ROWS_PER_WG * (DY / 2); idx += NTHREADS) {
        int r = idx / (DY / 2), c = idx % (DY / 2);
        float2 v = *(const float2*)(y + (size_t)(row0 + r) * DY + c * 2);
        yt[r * YPAD + c * 2] = v.x;
        yt[r * YPAD + c * 2 + 1] = v.y;
    }
#endif
    __syncthreads();

    // WMMA f32 16x16x4 fragment coordinates for this lane
    const int arow = lane & 15;          // A row / B column index
    const int kb   = (lane >> 4) << 1;   // K sub-offset {0,2}

    // online-logsumexp state: 2 M-blocks x 8 C-rows per lane
    float m_xy[NBLK][8], s_xy[NBLK][8];
    float m_x[NBLK][8],  s_x[NBLK][8];
    float m_y[NBLK][8],  s_y[NBLK][8];
#pragma unroll
    for (int b = 0; b < NBLK; ++b)
#pragma unroll
        for (int r = 0; r < 8; ++r) {
            m_xy[b][r] = m_x[b][r] = m_y[b][r] = -3.0e38f;
            s_xy[b][r] = s_x[b][r] = s_y[b][r] = 0.0f;
        }

    for (int jt = wave; jt < NS / 16; jt += NWAVE) {
        const int j0 = jt * 16;
        const float* xsr = xs + (size_t)(j0 + arow) * DX;
        const float* ysr = ys + (size_t)(j0 + arow) * DY;

        // ---- x dots: each B fragment feeds both M blocks ----
        v8f dx0 = {}, dx1 = {};
        float nx = 0.0f;
#pragma unroll 4
        for (int k0 = 0; k0 < DX; k0 += 4) {
            v2f b  = *(const v2f*)(xsr + k0 + kb);
            nx = fmaf(b.x, b.x, nx);
            nx = fmaf(b.y, b.y, nx);
            v2f a0 = *(const v2f*)(&xt[arow * XPAD + k0 + kb]);
            v2f a1 = *(const v2f*)(&xt[(arow + 16) * XPAD + k0 + kb]);
            dx0 = __builtin_amdgcn_wmma_f32_16x16x4_f32(false, a0, false, b, (short)0, dx0, false, false);
            dx1 = __builtin_amdgcn_wmma_f32_16x16x4_f32(false, a1, false, b, (short)0, dx1, false, false);
        }
        nx += __shfl_xor(nx, 16);        // full ||xs_col||^2

        // ---- y dots ----
        v8f dy0 = {}, dy1 = {};
        float ny = 0.0f;
#pragma unroll 4
        for (int k0 = 0; k0 < DY; k0 += 4) {
            v2f b  = *(const v2f*)(ysr + k0 + kb);
            ny = fmaf(b.x, b.x, ny);
            ny = fmaf(b.y, b.y, ny);
            v2f a0 = *(const v2f*)(&yt[arow * YPAD + k0 + kb]);
            v2f a1 = *(const v2f*)(&yt[(arow + 16) * YPAD + k0 + kb]);
            dy0 = __builtin_amdgcn_wmma_f32_16x16x4_f32(false, a0, false, b, (short)0, dy0, false, false);
            dy1 = __builtin_amdgcn_wmma_f32_16x16x4_f32(false, a1, false, b, (short)0, dy1, false, false);
        }
        ny += __shfl_xor(ny, 16);

        const float hx = 0.5f * nx;
        const float hy = 0.5f * ny;
#pragma unroll
        for (int r = 0; r < 8; ++r) {
            float px0 = dx0[r] - hx, py0 = dy0[r] - hy;
            lse_update(m_x[0][r],  s_x[0][r],  px0);
            lse_update(m_y[0][r],  s_y[0][r],  py0);
            lse_update(m_xy[0][r], s_xy[0][r], px0 + py0);
            float px1 = dx1[r] - hx, py1 = dy1[r] - hy;
            lse_update(m_x[1][r],  s_x[1][r],  px1);
            lse_update(m_y[1][r],  s_y[1][r],  py1);
            lse_update(m_xy[1][r], s_xy[1][r], px1 + py1);
        }
    }

    // ---- butterfly-combine the 16 sample-columns within each half-wave ----
#pragma unroll
    for (int b = 0; b < NBLK; ++b)
#pragma unroll
        for (int r = 0; r < 8; ++r)
            for (int mask = 1; mask <= 8; mask <<= 1) {
                lse_combine(m_x[b][r],  s_x[b][r],  __shfl_xor(m_x[b][r],  mask), __shfl_xor(s_x[b][r],  mask));
                lse_combine(m_y[b][r],  s_y[b][r],  __shfl_xor(m_y[b][r],  mask), __shfl_xor(s_y[b][r],  mask));
                lse_combine(m_xy[b][r], s_xy[b][r], __shfl_xor(m_xy[b][r], mask), __shfl_xor(s_xy[b][r], mask));
            }

    if ((lane & 15) == 0) {
        int rbase = (lane >> 4) * 8;
#pragma unroll
        for (int b = 0; b < NBLK; ++b)
#pragma unroll
            for (int r = 0; r < 8; ++r) {
                int row = b * 16 + rbase + r;
                red[wave][row][0][0] = m_xy[b][r]; red[wave][row][0][1] = s_xy[b][r];
                red[wave][row][1][0] = m_x[b][r];  red[wave][row][1][1] = s_x[b][r];
                red[wave][row][2][0] = m_y[b][r];  red[wave][row][2][1] = s_y[b][r];
            }
    }
    __syncthreads();

    // ---- merge 8 wave partials per row, z_row = lse_xy - lse_x - lse_y ----
    if (tid < ROWS_PER_WG) {
        float acc = 0.0f;
#pragma unroll
        for (int t = 0; t < 3; ++t) {
            float M = red[0][tid][t][0];
            float S = red[0][tid][t][1];
            for (int w = 1; w < NWAVE; ++w)
                lse_combine(M, S, red[w][tid][t][0], red[w][tid][t][1]);
            float l = M + __logf(S);
            acc += (t == 0) ? l : -l;
        }
        zrow[tid] = acc;
    }
    __syncthreads();

    if (tid == 0) {
        float s = 0.0f;
#pragma unroll
        for (int r = 0; r < ROWS_PER_WG; ++r) s += zrow[r];
        partial[blockIdx.x] = s;
    }
}

__global__ __launch_bounds__(NPART)
void mi_finalize_kernel(const float* __restrict__ partial, float* __restrict__ out) {
    __shared__ float sh[NPART];
    int t = threadIdx.x;
    sh[t] = partial[t];
    __syncthreads();
    for (int off = NPART / 2; off > 0; off >>= 1) {
        if (t < off) sh[t] += sh[t + off];
        __syncthreads();
    }
    if (t == 0) out[0] = sh[0] * (1.0f / (float)BATCH);
}

extern "C" void kernel_launch(void* const* d_in, const int* in_sizes, int n_in,
                              void* d_out, int out_size, void* d_ws, size_t ws_size,
                              hipStream_t stream) {
    const float* x  = (const float*)d_in[0];  // (4096, 256)
    const float* y  = (const float*)d_in[1];  // (4096, 128)
    const float* xs = (const float*)d_in[2];  // (8192, 256)
    const float* ys = (const float*)d_in[3];  // (8192, 128)
    float* partial  = (float*)d_ws;           // NPART floats

    mi_main_kernel<<<BATCH / ROWS_PER_WG, NTHREADS, 0, stream>>>(x, y, xs, ys, partial);
    mi_finalize_kernel<<<1, NPART, 0, stream>>>(partial, (float*)d_out);
}